// ShapeClassifier_36661840839572
// MI455X (gfx1250) — compile-verified
//
#include <hip/hip_runtime.h>
#include <math.h>

// ---------------------------------------------------------------------------
// E(3)-equivariant GNN, MI455X (gfx1250, wave32).
// Design: L2-resident node state + atomics; per-layer fused edge kernels.
// Radial-MLP layer 1 runs on V_WMMA_F32_16X16X4_F32 (16-edge tiles, K=8 as
// two K=4 chunks, 2 heads packed across N). Everything else is small VALU.
// ---------------------------------------------------------------------------

#define NODES_TOTAL (4096 * 32)
#define EDGES_TOTAL (NODES_TOTAL * 16)
#define GRAPHS_TOTAL 4096
#define WAVES_PER_BLOCK 8
#define EDGES_PER_WAVE 16
#define EDGES_PER_BLOCK (WAVES_PER_BLOCK * EDGES_PER_WAVE)
#define INV_SQRT3 0.57735026918962576451f
#define INV_SQRT2 0.70710678118654752440f

typedef float v2f __attribute__((ext_vector_type(2)));
typedef float v8f __attribute__((ext_vector_type(8)));

__device__ __forceinline__ float sigmoid_f(float x) {
  return 1.0f / (1.0f + __expf(-x));
}

// ---------------------------------------------------------------------------
// Fused per-layer edge kernel.
// Wave tile = 16 edges. Phase 1: lanes 0-15 do geometry + RBF -> LDS.
// Phase 2: WMMA f32 16x16x4 (x2 K-chunks) per head-pair: h = relu(rbf@W1+b1),
//          C layout gives lane = hidden column, VGPR = edge row -> LDS.
// Phase 3: lanes split into 2 groups over the same 16 edges; each group
//          finishes its heads' radials (8->4 VALU) and scatters messages
//          with global_atomic_add_f32.
// LAYER=1 heads {0,1}; LAYER=2 heads {2..6}; LAYER=3 heads {7,8}.
// ---------------------------------------------------------------------------
template <int LAYER>
__global__ __launch_bounds__(256) void edge_layer_kernel(
    const float* __restrict__ positions, const float* __restrict__ node_feat,
    const int* __restrict__ senders, const int* __restrict__ receivers,
    const float* __restrict__ rad_w1, const float* __restrict__ rad_b1,
    const float* __restrict__ rad_w2, const float* __restrict__ rad_b2,
    const float* __restrict__ w_si1,
    const float* __restrict__ featsIn0, const float* __restrict__ featsIn1,
    float* __restrict__ accA, float* __restrict__ accB,
    float* __restrict__ accC, float* __restrict__ accD,
    float* __restrict__ accE) {
  constexpr int NH = (LAYER == 2) ? 5 : 2;
  constexpr int H0 = (LAYER == 1) ? 0 : ((LAYER == 2) ? 2 : 7);
  constexpr int NPAIR = (NH + 1) / 2;

  __shared__ float s_rbf[WAVES_PER_BLOCK][16][8];
  __shared__ float s_u[WAVES_PER_BLOCK][16][4];
  __shared__ float s_h[WAVES_PER_BLOCK][16][NH * 8];

  const int w = threadIdx.x >> 5;
  const int lane = threadIdx.x & 31;
  const int l16 = lane & 15;
  const int hi = lane >> 4;  // half-wave id
  const int eBase = blockIdx.x * EDGES_PER_BLOCK + w * EDGES_PER_WAVE;

  // ---- Phase 1: geometry + RBF (one edge per lane, lanes 0-15) ----
  if (lane < 16) {
    const int e = eBase + lane;
    const int s = senders[e];
    const int r = receivers[e];
    const float dx = positions[3 * r + 0] - positions[3 * s + 0];
    const float dy = positions[3 * r + 1] - positions[3 * s + 1];
    const float dz = positions[3 * r + 2] - positions[3 * s + 2];
    const float dist = sqrtf(dx * dx + dy * dy + dz * dz);
    const float inv = 1.0f / fmaxf(dist, 1e-9f);
    s_u[w][lane][0] = dx * inv;
    s_u[w][lane][1] = dy * inv;
    s_u[w][lane][2] = dz * inv;
#pragma unroll
    for (int k = 0; k < 8; ++k) {  // centers = linspace(0,3.5,8) -> 0.5*k
      const float t = dist - 0.5f * (float)k;
      s_rbf[w][lane][k] = __expf(-4.0f * t * t);
    }
  }
  __syncthreads();

  // ---- Phase 2: radial layer-1 via WMMA f32 16x16x4 ----
  // A (16x4 f32): lanes 0-15 hold K=0,1 / lanes 16-31 hold K=2,3; M = lane&15.
  v2f a0, a1;
  a0.x = s_rbf[w][l16][2 * hi + 0];
  a0.y = s_rbf[w][l16][2 * hi + 1];
  a1.x = s_rbf[w][l16][2 * hi + 4];
  a1.y = s_rbf[w][l16][2 * hi + 5];

#pragma unroll
  for (int pr = 0; pr < NPAIR; ++pr) {
    const int hl = 2 * pr + (l16 >> 3);  // head-local index of this column
    const int j = l16 & 7;               // hidden index within head
    const bool valid = (hl < NH);
    const int head = H0 + (valid ? hl : 0);
    // B (4x16 f32), mirrored layout: lanes 0-15 hold K=0,1 / 16-31 hold K=2,3.
    v2f b0 = {0.f, 0.f}, b1 = {0.f, 0.f};
    if (valid) {
      const float* W = rad_w1 + head * 64;  // [k*8 + j]
      b0.x = W[(2 * hi + 0) * 8 + j];
      b0.y = W[(2 * hi + 1) * 8 + j];
      b1.x = W[(2 * hi + 4) * 8 + j];
      b1.y = W[(2 * hi + 5) * 8 + j];
    }
    v8f c = {0.f, 0.f, 0.f, 0.f, 0.f, 0.f, 0.f, 0.f};
    c = __builtin_amdgcn_wmma_f32_16x16x4_f32(false, a0, false, b0, (short)0,
                                              c, false, false);
    c = __builtin_amdgcn_wmma_f32_16x16x4_f32(false, a1, false, b1, (short)0,
                                              c, false, false);
    if (valid) {
      const float bias = rad_b1[head * 8 + j];
#pragma unroll
      for (int v = 0; v < 8; ++v) {  // C: lane = column, VGPR v -> edge v+8*hi
        s_h[w][v + 8 * hi][hl * 8 + j] = fmaxf(c[v] + bias, 0.0f);
      }
    }
  }
  __syncthreads();

  // ---- Phase 3: radial layer-2 + messages + atomic scatter ----
  const int e = l16;
  const int eg = eBase + e;
  const int s = senders[eg];
  const int r = receivers[eg];
  const float ux = s_u[w][e][0], uy = s_u[w][e][1], uz = s_u[w][e][2];
  const float* hrow = &s_h[w][e][0];

  auto radial = [&](int head, float* R) {
    const int hl = head - H0;
    const float* W2 = rad_w2 + head * 32;  // [j*4 + c]
#pragma unroll
    for (int cc = 0; cc < 4; ++cc) {
      float acc = rad_b2[head * 4 + cc];
#pragma unroll
      for (int jj = 0; jj < 8; ++jj)
        acc = fmaf(hrow[hl * 8 + jj], W2[jj * 4 + cc], acc);
      R[cc] = acc;
    }
  };

  if (LAYER == 1) {
    const float x0 = node_feat[s];
    if (hi == 0) {  // f0 += R0 * x0s
      float R[4];
      radial(0, R);
#pragma unroll
      for (int c = 0; c < 4; ++c)
        atomicAdd(&accA[r * 4 + c], R[c] * w_si1[c] * x0);
    } else {  // f1 += R1 * x0s * u
      float R[4];
      radial(1, R);
#pragma unroll
      for (int c = 0; c < 4; ++c) {
        const float m = R[c] * w_si1[c] * x0;
        atomicAdd(&accB[r * 12 + c * 3 + 0], m * ux);
        atomicAdd(&accB[r * 12 + c * 3 + 1], m * uy);
        atomicAdd(&accB[r * 12 + c * 3 + 2], m * uz);
      }
    }
  } else if (LAYER == 2) {
    if (hi == 0) {  // p00_0, p01_1 (scalar-path features)
      float x0s[4];
#pragma unroll
      for (int c = 0; c < 4; ++c) x0s[c] = featsIn0[s * 4 + c];
      float R2[4], R3[4];
      radial(2, R2);
      radial(3, R3);
#pragma unroll
      for (int c = 0; c < 4; ++c) {
        atomicAdd(&accA[r * 4 + c], R2[c] * x0s[c]);
        const float m = R3[c] * x0s[c];
        atomicAdd(&accB[r * 12 + c * 3 + 0], m * ux);
        atomicAdd(&accB[r * 12 + c * 3 + 1], m * uy);
        atomicAdd(&accB[r * 12 + c * 3 + 2], m * uz);
      }
    } else {  // p10_1, p11_0 (dot), p11_1 (cross) (vector-path features)
      float x1s[4][3];
#pragma unroll
      for (int c = 0; c < 4; ++c) {
        x1s[c][0] = featsIn1[s * 12 + c * 3 + 0];
        x1s[c][1] = featsIn1[s * 12 + c * 3 + 1];
        x1s[c][2] = featsIn1[s * 12 + c * 3 + 2];
      }
      float R4[4], R5[4], R6[4];
      radial(4, R4);
      radial(5, R5);
      radial(6, R6);
#pragma unroll
      for (int c = 0; c < 4; ++c) {
        atomicAdd(&accC[r * 12 + c * 3 + 0], R4[c] * x1s[c][0]);
        atomicAdd(&accC[r * 12 + c * 3 + 1], R4[c] * x1s[c][1]);
        atomicAdd(&accC[r * 12 + c * 3 + 2], R4[c] * x1s[c][2]);
        const float dot = x1s[c][0] * ux + x1s[c][1] * uy + x1s[c][2] * uz;
        atomicAdd(&accD[r * 4 + c], R5[c] * dot * INV_SQRT3);
        const float cr0 = x1s[c][1] * uz - x1s[c][2] * uy;
        const float cr1 = x1s[c][2] * ux - x1s[c][0] * uz;
        const float cr2 = x1s[c][0] * uy - x1s[c][1] * ux;
        atomicAdd(&accE[r * 12 + c * 3 + 0], R6[c] * cr0 * INV_SQRT2);
        atomicAdd(&accE[r * 12 + c * 3 + 1], R6[c] * cr1 * INV_SQRT2);
        atomicAdd(&accE[r * 12 + c * 3 + 2], R6[c] * cr2 * INV_SQRT2);
      }
    }
  } else {  // LAYER == 3
    if (hi == 0) {  // q00 += R7 * x0s
      float R[4];
      radial(7, R);
#pragma unroll
      for (int c = 0; c < 4; ++c)
        atomicAdd(&accA[r * 4 + c], R[c] * featsIn0[s * 4 + c]);
    } else {  // q11 += R8 * dot(x1s, u) / sqrt(3)
      float R[4];
      radial(8, R);
#pragma unroll
      for (int c = 0; c < 4; ++c) {
        const float dot = featsIn1[s * 12 + c * 3 + 0] * ux +
                          featsIn1[s * 12 + c * 3 + 1] * uy +
                          featsIn1[s * 12 + c * 3 + 2] * uz;
        atomicAdd(&accB[r * 4 + c], R[c] * dot * INV_SQRT3);
      }
    }
  }
}

// ---- Node kernel 1: self-interact (4x4) + gate after layer 1 ----
__global__ __launch_bounds__(256) void node_gate1_kernel(
    const float* __restrict__ f0, const float* __restrict__ f1,
    const float* __restrict__ w20, const float* __restrict__ w21,
    float* __restrict__ feats0, float* __restrict__ feats1) {
  const int n = blockIdx.x * blockDim.x + threadIdx.x;
  if (n >= NODES_TOTAL) return;
  float a[4], b[4][3];
#pragma unroll
  for (int i = 0; i < 4; ++i) a[i] = f0[n * 4 + i];
#pragma unroll
  for (int i = 0; i < 4; ++i) {
    b[i][0] = f1[n * 12 + i * 3 + 0];
    b[i][1] = f1[n * 12 + i * 3 + 1];
    b[i][2] = f1[n * 12 + i * 3 + 2];
  }
#pragma unroll
  for (int o = 0; o < 4; ++o) {
    float y = 0.f, y0 = 0.f, y1 = 0.f, y2 = 0.f;
#pragma unroll
    for (int i = 0; i < 4; ++i) {
      y = fmaf(w20[o * 4 + i], a[i], y);
      y0 = fmaf(w21[o * 4 + i], b[i][0], y0);
      y1 = fmaf(w21[o * 4 + i], b[i][1], y1);
      y2 = fmaf(w21[o * 4 + i], b[i][2], y2);
    }
    feats0[n * 4 + o] = y * sigmoid_f(y);  // silu
    const float sg = sigmoid_f(sqrtf(y0 * y0 + y1 * y1 + y2 * y2));
    feats1[n * 12 + o * 3 + 0] = y0 * sg;
    feats1[n * 12 + o * 3 + 1] = y1 * sg;
    feats1[n * 12 + o * 3 + 2] = y2 * sg;
  }
}

// ---- Node kernel 2: concat + self-interact (4x8, 4x12) + gate ----
__global__ __launch_bounds__(256) void node_gate2_kernel(
    const float* __restrict__ p00, const float* __restrict__ p01,
    const float* __restrict__ p10, const float* __restrict__ p11_0,
    const float* __restrict__ p11_1, const float* __restrict__ w30,
    const float* __restrict__ w31, float* __restrict__ feats0,
    float* __restrict__ feats1) {
  const int n = blockIdx.x * blockDim.x + threadIdx.x;
  if (n >= NODES_TOTAL) return;
  float f0c[8], f1c[12][3];
#pragma unroll
  for (int i = 0; i < 4; ++i) {
    f0c[i] = p00[n * 4 + i];
    f0c[4 + i] = p11_0[n * 4 + i];
#pragma unroll
    for (int d = 0; d < 3; ++d) {
      f1c[i][d] = p01[n * 12 + i * 3 + d];
      f1c[4 + i][d] = p10[n * 12 + i * 3 + d];
      f1c[8 + i][d] = p11_1[n * 12 + i * 3 + d];
    }
  }
#pragma unroll
  for (int o = 0; o < 4; ++o) {
    float y = 0.f;
#pragma unroll
    for (int i = 0; i < 8; ++i) y = fmaf(w30[o * 8 + i], f0c[i], y);
    feats0[n * 4 + o] = y * sigmoid_f(y);
    float y0 = 0.f, y1 = 0.f, y2 = 0.f;
#pragma unroll
    for (int i = 0; i < 12; ++i) {
      y0 = fmaf(w31[o * 12 + i], f1c[i][0], y0);
      y1 = fmaf(w31[o * 12 + i], f1c[i][1], y1);
      y2 = fmaf(w31[o * 12 + i], f1c[i][2], y2);
    }
    const float sg = sigmoid_f(sqrtf(y0 * y0 + y1 * y1 + y2 * y2));
    feats1[n * 12 + o * 3 + 0] = y0 * sg;
    feats1[n * 12 + o * 3 + 1] = y1 * sg;
    feats1[n * 12 + o * 3 + 2] = y2 * sg;
  }
}

// ---- Final: self-interact (4x8) + silu + per-graph wave32 reduce + w_out ----
__global__ __launch_bounds__(32) void finish_kernel(
    const float* __restrict__ q00, const float* __restrict__ q11,
    const float* __restrict__ w_si4, const float* __restrict__ w_out,
    float* __restrict__ out) {
  const int g = blockIdx.x;
  const int lane = threadIdx.x;  // one node per lane, 32 nodes per graph
  const int n = g * 32 + lane;
  float f[8];
#pragma unroll
  for (int c = 0; c < 4; ++c) {
    f[c] = q00[n * 4 + c];
    f[4 + c] = q11[n * 4 + c];
  }
  float v[4];
#pragma unroll
  for (int o = 0; o < 4; ++o) {
    float y = 0.f;
#pragma unroll
    for (int i = 0; i < 8; ++i) y = fmaf(w_si4[o * 8 + i], f[i], y);
    v[o] = y * sigmoid_f(y);
  }
  // wave32 butterfly reduction over nodes of this graph
#pragma unroll
  for (int m = 16; m >= 1; m >>= 1) {
#pragma unroll
    for (int o = 0; o < 4; ++o) v[o] += __shfl_xor(v[o], m, 32);
  }
  if (lane < 8) {
    float acc = 0.f;
#pragma unroll
    for (int c = 0; c < 4; ++c) acc = fmaf(w_out[lane * 4 + c], v[c], acc);
    out[g * 8 + lane] = acc;
  }
}

extern "C" void kernel_launch(void* const* d_in, const int* in_sizes, int n_in,
                              void* d_out, int out_size, void* d_ws,
                              size_t ws_size, hipStream_t stream) {
  (void)in_sizes; (void)n_in; (void)out_size; (void)ws_size;
  const float* positions = (const float*)d_in[0];
  const float* node_feat = (const float*)d_in[1];
  const float* w_si1 = (const float*)d_in[2];
  const float* w_si2_0 = (const float*)d_in[3];
  const float* w_si2_1 = (const float*)d_in[4];
  const float* w_si3_0 = (const float*)d_in[5];
  const float* w_si3_1 = (const float*)d_in[6];
  const float* w_si4 = (const float*)d_in[7];
  const float* w_out = (const float*)d_in[8];
  const float* rad_w1 = (const float*)d_in[9];
  const float* rad_b1 = (const float*)d_in[10];
  const float* rad_w2 = (const float*)d_in[11];
  const float* rad_b2 = (const float*)d_in[12];
  const int* senders = (const int*)d_in[13];
  const int* receivers = (const int*)d_in[14];

  float* W = (float*)d_ws;
  const size_t N = NODES_TOTAL;
  float* acc1_f0 = W + 0 * N;    // 4N
  float* acc1_f1 = W + 4 * N;    // 12N
  float* p00     = W + 16 * N;   // 4N
  float* p01     = W + 20 * N;   // 12N
  float* p10     = W + 32 * N;   // 12N
  float* p11_0   = W + 44 * N;   // 4N
  float* p11_1   = W + 48 * N;   // 12N
  float* q00     = W + 60 * N;   // 4N
  float* q11     = W + 64 * N;   // 4N  (accumulators end at 68N)
  float* feats0  = W + 68 * N;   // 4N
  float* feats1  = W + 72 * N;   // 12N (total 84N floats ~ 42 MB)

  // zero all atomic accumulators (graph-capture safe)
  hipMemsetAsync(W, 0, sizeof(float) * 68 * N, stream);

  const dim3 eb(256);
  const dim3 eg(EDGES_TOTAL / EDGES_PER_BLOCK);
  const dim3 nb(256);
  const dim3 ng(NODES_TOTAL / 256);

  edge_layer_kernel<1><<<eg, eb, 0, stream>>>(
      positions, node_feat, senders, receivers, rad_w1, rad_b1, rad_w2, rad_b2,
      w_si1, nullptr, nullptr, acc1_f0, acc1_f1, nullptr, nullptr, nullptr);
  node_gate1_kernel<<<ng, nb, 0, stream>>>(acc1_f0, acc1_f1, w_si2_0, w_si2_1,
                                           feats0, feats1);
  edge_layer_kernel<2><<<eg, eb, 0, stream>>>(
      positions, node_feat, senders, receivers, rad_w1, rad_b1, rad_w2, rad_b2,
      w_si1, feats0, feats1, p00, p01, p10, p11_0, p11_1);
  node_gate2_kernel<<<ng, nb, 0, stream>>>(p00, p01, p10, p11_0, p11_1,
                                           w_si3_0, w_si3_1, feats0, feats1);
  edge_layer_kernel<3><<<eg, eb, 0, stream>>>(
      positions, node_feat, senders, receivers, rad_w1, rad_b1, rad_w2, rad_b2,
      w_si1, feats0, feats1, q00, q11, nullptr, nullptr, nullptr);
  finish_kernel<<<GRAPHS_TOTAL, 32, 0, stream>>>(q00, q11, w_si4, w_out,
                                                 (float*)d_out);
}